// GaussianRenderer_68564857913435
// MI455X (gfx1250) — compile-verified
//
#include <hip/hip_runtime.h>

typedef float v2f __attribute__((ext_vector_type(2)));
typedef float v8f __attribute__((ext_vector_type(8)));

#define N_G   4096
#define IMGH  512
#define REC_STR 16   // ws floats per gaussian (64B records)
#define LREC  13     // LDS floats per compacted record (odd -> no bank conflicts)
#define CAP   1024   // compacted-record ring capacity

// ---------------------------------------------------------------------------
// Kernel 1: stable depth-rank sort + per-gaussian preprocessing, scatter into
// sorted order in workspace.
// ws record: [px, py, radius, 0, f0..f5, op, cr, cg, cb, 0, 0]
// f = [ia, id, B, -(2 ia px + B py), -(2 id py + B px), ia px^2 + id py^2 + B px py]
// ---------------------------------------------------------------------------
__global__ __launch_bounds__(256) void prep_kernel(
    const float* __restrict__ pos2d, const float* __restrict__ cov2d,
    const float* __restrict__ opacity, const float* __restrict__ color,
    float* __restrict__ ws)
{
  __shared__ float sdepth[N_G];
  const int tid = threadIdx.x;
  for (int k = tid; k < N_G; k += 256) sdepth[k] = pos2d[k * 3 + 2];
  __syncthreads();

  const int i = blockIdx.x * 256 + tid;
  const float di = sdepth[i];
  int rank = 0;
  for (int j = 0; j < N_G; ++j) {
    float dj = sdepth[j];
    rank += (dj < di) || (dj == di && j < i);   // stable argsort semantics
  }

  const float px = pos2d[i * 3 + 0];
  const float py = pos2d[i * 3 + 1];
  const float a = cov2d[i * 4 + 0], b = cov2d[i * 4 + 1];
  const float c = cov2d[i * 4 + 2], d = cov2d[i * 4 + 3];
  const float tr  = a + d;
  const float det = a * d - b * c;
  const float term2 = 0.5f * sqrtf(fmaxf(tr * tr - 4.0f * det, 0.0f));
  const float e1 = 0.5f * tr - term2, e2 = 0.5f * tr + term2;
  const float radius = 3.0f * sqrtf(fmaxf(e1, e2));
  const float invdet = 1.0f / det;
  const float ia  = d * invdet;
  const float id_ = a * invdet;
  const float Bc  = -(b + c) * invdet;            // ib + ic
  const float gx  = -(2.0f * ia * px + Bc * py);
  const float gy  = -(2.0f * id_ * py + Bc * px);
  const float g1  = ia * px * px + id_ * py * py + Bc * px * py;

  const float op = opacity[i];
  const float cr = fmaxf(color[i * 3 + 0] + 0.5f, 0.0f);
  const float cg = fmaxf(color[i * 3 + 1] + 0.5f, 0.0f);
  const float cb = fmaxf(color[i * 3 + 2] + 0.5f, 0.0f);

  float* r = ws + (size_t)rank * REC_STR;
  r[0] = px;  r[1] = py;  r[2] = radius; r[3]  = 0.0f;
  r[4] = ia;  r[5] = id_; r[6] = Bc;     r[7]  = gx;
  r[8] = gy;  r[9] = g1;  r[10] = op;    r[11] = cr;
  r[12] = cg; r[13] = cb; r[14] = 0.0f;  r[15] = 0.0f;
}

// ---------------------------------------------------------------------------
// Kernel 2: per-tile ordered compaction + WMMA-based evaluation + compositing.
// Grid: 2048 blocks x 256 threads. Block = 8x16 pixel patch of one 64x64 tile.
// Wave w owns pixel column x (fixed), 16 y's (one per lane%16).
// q(16 gaussians x 16 pixels) computed with two chained V_WMMA_F32_16X16X4_F32.
// ---------------------------------------------------------------------------
__global__ __launch_bounds__(256) void render_kernel(
    const float* __restrict__ ws, float* __restrict__ out)
{
  __shared__ float rec[(CAP + 16) * LREC];
  __shared__ int waveCnt[8];

  const int tid  = threadIdx.x;
  const int lane = tid & 31;
  const int w    = tid >> 5;
  const int half = lane >> 4;      // 0: K0,K1,K4,K5 / gaussians M=r ; 1: K2,K3,K6,K7 / M=r+8
  const int row  = lane & 15;      // pixel index N within the wave

  const int b    = blockIdx.x;
  const int tile = b >> 5, sub = b & 31;
  const int tx = tile >> 3, ty = tile & 7;
  const int sx = sub >> 2,  sy = sub & 3;
  const float left = (float)(tx * 64);
  const float top  = (float)(ty * 64);
  const int x  = tx * 64 + sx * 8 + w;
  const int y0 = ty * 64 + sy * 16;

  const float fx = (float)x;
  const float fy = (float)(y0 + row);

  // Pixel feature matrix B (4x16 f32, two WMMAs for K=0..7):
  // F = [x^2, y^2, x*y, x, y, 1, 0, 0]; lane half selects K vs K+2 per VGPR.
  v2f B1, B2;
  B1.x = half ? fx * fy : fx * fx;   // K2 : K0
  B1.y = half ? fx      : fy * fy;   // K3 : K1
  B2.x = half ? 0.0f    : fy;        // K6 : K4
  B2.y = half ? 0.0f    : 1.0f;      // K7 : K5

  float T = 1.0f, accR = 0.0f, accG = 0.0f, accB = 0.0f;
  int fill = 0;

  for (int base = 0; base < N_G; base += 256) {
    // ---- ordered compaction round (256 sorted gaussians) ----
    const int i = base + tid;
    if (base + 256 < N_G)
      __builtin_prefetch(ws + (size_t)(i + 256) * REC_STR, 0, 1);

    const float4 hdr = *(const float4*)(ws + (size_t)i * REC_STR); // px,py,rad,0
    const bool pred = (hdr.x + hdr.z >= left) && (hdr.x - hdr.z < left + 64.0f) &&
                      (hdr.y + hdr.z >= top)  && (hdr.y - hdr.z < top  + 64.0f);

    const unsigned mask = (unsigned)__ballot(pred);
    const int prefix = __popc(mask & ((1u << lane) - 1u));
    if (lane == 0) waveCnt[w] = __popc(mask);
    __syncthreads();

    int off = 0, total = 0;
    for (int v = 0; v < 8; ++v) {
      int cN = waveCnt[v];
      if (v < w) off += cN;
      total += cN;
    }
    if (pred) {
      const int pos = fill + off + prefix;       // order preserved globally
      const float* src = ws + (size_t)i * REC_STR;
      float* dst = rec + pos * LREC;
      dst[0] = src[4]; dst[1] = src[5]; dst[2]  = src[6];  dst[3]  = src[7];
      dst[4] = src[8]; dst[5] = src[9]; dst[6]  = 0.0f;    dst[7]  = 0.0f;
      dst[8] = src[10]; dst[9] = src[11]; dst[10] = src[12]; dst[11] = src[13];
    }
    fill += total;
    __syncthreads();

    // ---- flush: composite everything in the ring, in sorted order ----
    if (fill > CAP - 256 || base + 256 >= N_G) {
      for (int c0 = 0; c0 < fill; c0 += 16) {
        // A matrix: 16 gaussians x K (f32 16x4 layout, rows striped by lane%16)
        const int rbase = (c0 + row) * LREC;
        v2f A1, A2;
        A1.x = rec[rbase + (half ? 2 : 0)];
        A1.y = rec[rbase + (half ? 3 : 1)];
        A2.x = rec[rbase + (half ? 6 : 4)];
        A2.y = rec[rbase + (half ? 7 : 5)];

        v8f q = {};
        q = __builtin_amdgcn_wmma_f32_16x16x4_f32(false, A1, false, B1,
                                                  (short)0, q, false, false);
        q = __builtin_amdgcn_wmma_f32_16x16x4_f32(false, A2, false, B2,
                                                  (short)0, q, false, false);

        // Phase 1: 8 independent alphas (exp pipelines freely, branchless tail
        // handling via cndmask — keeps EXEC full, no divergent blocks).
        const int mb   = c0 + (half << 3);       // first gaussian of this half
        const int vlim = fill - mb;              // #valid among my 8
        float alpha[8];
#pragma unroll
        for (int r = 0; r < 8; ++r) {
          const float prob = __builtin_amdgcn_exp2f(q[r] * -0.72134752f); // e^{-q/2}
          const float op   = rec[(mb + r) * LREC + 8];
          const float av   = fminf(fmaxf(op * prob, 0.01f), 0.99f);
          alpha[r] = (r < vlim) ? av : 0.0f;
        }

        // Phase 2: pure-FMA sequential compositing scan over this lane's 8.
        float tl = 1.0f, sr = 0.0f, sg = 0.0f, sb = 0.0f;
#pragma unroll
        for (int r = 0; r < 8; ++r) {
          const float wgt = alpha[r] * tl;
          sr += wgt * rec[(mb + r) * LREC + 9];
          sg += wgt * rec[(mb + r) * LREC + 10];
          sb += wgt * rec[(mb + r) * LREC + 11];
          tl *= (1.0f - alpha[r]);
        }

        // stitch the two lane-halves (gaussians 0..7 | 8..15) in order
        const float to  = __shfl_xor(tl, 16, 32);
        const float sro = __shfl_xor(sr, 16, 32);
        const float sgo = __shfl_xor(sg, 16, 32);
        const float sbo = __shfl_xor(sb, 16, 32);
        const float t_lo  = half ? to  : tl,  t_hi  = half ? tl : to;
        const float sr_lo = half ? sro : sr,  sr_hi = half ? sr : sro;
        const float sg_lo = half ? sgo : sg,  sg_hi = half ? sg : sgo;
        const float sb_lo = half ? sbo : sb,  sb_hi = half ? sb : sbo;

        accR += T * (sr_lo + t_lo * sr_hi);
        accG += T * (sg_lo + t_lo * sg_hi);
        accB += T * (sb_lo + t_lo * sb_hi);
        T *= t_lo * t_hi;
      }
      __syncthreads();
      fill = 0;
    }
  }

  if (half == 0) {   // both halves hold identical accumulators; one writes
    const int y = y0 + row;
    float* o = out + ((size_t)x * IMGH + y) * 3;
    o[0] = accR; o[1] = accG; o[2] = accB;
  }
}

// ---------------------------------------------------------------------------
extern "C" void kernel_launch(void* const* d_in, const int* in_sizes, int n_in,
                              void* d_out, int out_size, void* d_ws, size_t ws_size,
                              hipStream_t stream) {
  const float* pos2d   = (const float*)d_in[0];  // (N,3)
  const float* cov2d   = (const float*)d_in[1];  // (N,2,2)
  const float* opacity = (const float*)d_in[2];  // (N,)
  const float* color   = (const float*)d_in[3];  // (N,3)
  float* out = (float*)d_out;                    // (512,512,3)
  float* ws  = (float*)d_ws;                     // sorted records: N_G*16 floats

  prep_kernel<<<N_G / 256, 256, 0, stream>>>(pos2d, cov2d, opacity, color, ws);
  render_kernel<<<2048, 256, 0, stream>>>(ws, out);
}